// CTCLoss_21380347199935
// MI455X (gfx1250) — compile-verified
//
#include <hip/hip_runtime.h>

// Problem constants (from reference setup_inputs)
#define B_   64
#define T_   128
#define C_   6625
#define L_   25
#define S_   51          // 2*L+1
#define SPAD 64          // padded S for clean indexing
#define NEGV (-1e30f)

typedef __attribute__((ext_vector_type(2))) float v2f;
typedef __attribute__((ext_vector_type(8))) float v8f;

// ---------------------------------------------------------------------------
// Kernel 1: one block per (b,t) row. Single-pass online logsumexp over C,
// block-combine partials (max tree + WMMA f32 sum), then gather the 51
// extended-label log-probs into the workspace: lpext[row*SPAD + s].
// ---------------------------------------------------------------------------
__global__ __launch_bounds__(256) void k_rowlse(const float* __restrict__ x,
                                                const int* __restrict__ labels,
                                                float* __restrict__ lpext) {
    const int row = blockIdx.x;          // row = b*T + t
    const int b   = row / T_;
    const float* __restrict__ xr = x + (size_t)row * C_;
    const int tid = threadIdx.x;

    // per-thread online logsumexp over strided columns
    float m = -3.402823466e38f;
    float s = 0.0f;
    for (int c = tid; c < C_; c += 256) {
        float v  = xr[c];
        float mn = fmaxf(m, v);
        s = s * __expf(m - mn) + __expf(v - mn);
        m = mn;
    }

    __shared__ float sm[256];
    __shared__ float sv[256];
    __shared__ float sred;

    // block max
    sm[tid] = m;
    __syncthreads();
    for (int off = 128; off > 0; off >>= 1) {
        if (tid < off) sm[tid] = fmaxf(sm[tid], sm[tid + off]);
        __syncthreads();
    }
    const float M = sm[0];

    // rescale partial sums to common max
    sv[tid] = s * __expf(m - M);
    __syncthreads();

    // WMMA f32 reduction of the 256 partials. Executed redundantly by all 8
    // waves (EXEC all ones, zero divergence). A-tile layout per ISA 7.12.2:
    //   lane<16: VGPR0=A[m][0], VGPR1=A[m][1]; lane>=16: K=2,3.  m = lane&15.
    // B = all ones (layout-invariant). After 4 accumulating WMMAs,
    // D[m][*] = sum of 16 values of row m; per-lane sum of the 8 C VGPRs
    // gives rows 0-7 (lanes 0-15) / rows 8-15 (lanes 16-31); one xor-16
    // shuffle completes the 256-element sum.
    {
        const int lane  = tid & 31;
        const int mrow  = lane & 15;
        const int khalf = (lane >> 4) << 1;   // 0 or 2
        v8f acc = {};                          // C = 0
        v2f ones; ones.x = 1.0f; ones.y = 1.0f;
        for (int c = 0; c < 4; ++c) {
            const int base = c * 64 + mrow * 4 + khalf;
            v2f a; a.x = sv[base]; a.y = sv[base + 1];
            acc = __builtin_amdgcn_wmma_f32_16x16x4_f32(
                false, a, false, ones, (short)0, acc, false, false);
        }
        float p = acc[0] + acc[1] + acc[2] + acc[3] +
                  acc[4] + acc[5] + acc[6] + acc[7];
        float q = __shfl(p, lane ^ 16, 32);
        if (tid == 0) sred = M + __logf(p + q);   // row logsumexp
    }
    __syncthreads();
    const float lse = sred;

    // gather extended-label log-probs: ext[s] = blank(0) for even s,
    // labels[b][s>>1] for odd s
    if (tid < S_) {
        const int e = (tid & 1) ? labels[b * L_ + (tid >> 1)] : 0;
        lpext[(size_t)row * SPAD + tid] = xr[e] - lse;
    }
}

// ---------------------------------------------------------------------------
// Kernel 2: CTC forward recursion. One block per batch, one thread per state.
// ---------------------------------------------------------------------------
__global__ __launch_bounds__(64) void k_ctc(const float* __restrict__ lpext,
                                            const int* __restrict__ labels,
                                            const int* __restrict__ lens,
                                            float* __restrict__ losses) {
    const int b = blockIdx.x;
    const int s = threadIdx.x;
    const bool act = (s < S_);

    __shared__ float alpha[SPAD];

    int  ext = 0;
    bool skip = false;
    if (act && (s & 1)) {
        ext = labels[b * L_ + (s >> 1)];
        int extm2 = (s >= 3) ? labels[b * L_ + (s >> 1) - 1] : 0;
        skip = (ext != 0) && (ext != extm2);
    }

    const float* __restrict__ lp = lpext + (size_t)b * T_ * SPAD;

    // t = 0 init: alpha[0]=lp[0], alpha[1]=lp[1], else NEG
    alpha[s] = (act && s < 2) ? lp[s] : NEGV;
    __syncthreads();

    for (int t = 1; t < T_; ++t) {
        float a0 = alpha[s];
        float a1 = (s >= 1) ? alpha[s - 1] : NEGV;
        float a2 = (s >= 2 && skip) ? alpha[s - 2] : NEGV;
        float mm  = fmaxf(a0, fmaxf(a1, a2));
        float lse = mm + __logf(__expf(a0 - mm) + __expf(a1 - mm) + __expf(a2 - mm));
        float nv  = act ? (lse + lp[t * SPAD + s]) : NEGV;
        __syncthreads();
        alpha[s] = nv;
        __syncthreads();
    }

    if (s == 0) {
        const int il = 2 * lens[b];           // 2..50
        float aL = alpha[il];
        float aP = alpha[il - 1];
        float mm = fmaxf(aL, aP);
        float loss = -(mm + __logf(__expf(aL - mm) + __expf(aP - mm)));
        losses[b] = (loss < 1e29f) ? loss : 0.0f;   // zero_infinity
    }
}

// ---------------------------------------------------------------------------
// Kernel 3: deterministic mean of 64 losses.
// ---------------------------------------------------------------------------
__global__ __launch_bounds__(64) void k_mean(const float* __restrict__ losses,
                                             float* __restrict__ out) {
    __shared__ float buf[64];
    const int tid = threadIdx.x;
    buf[tid] = losses[tid];
    __syncthreads();
    for (int off = 32; off > 0; off >>= 1) {
        if (tid < off) buf[tid] += buf[tid + off];
        __syncthreads();
    }
    if (tid == 0) out[0] = buf[0] * (1.0f / 64.0f);
}

// ---------------------------------------------------------------------------
extern "C" void kernel_launch(void* const* d_in, const int* in_sizes, int n_in,
                              void* d_out, int out_size, void* d_ws, size_t ws_size,
                              hipStream_t stream) {
    const float* predicts = (const float*)d_in[0];   // [B,T,C] f32
    const int*   labels   = (const int*)d_in[1];     // [B,L]  int32
    const int*   lens     = (const int*)d_in[2];     // [B]    int32
    float* out = (float*)d_out;

    float* lpext  = (float*)d_ws;                          // B*T*SPAD floats (2 MB)
    float* losses = lpext + (size_t)B_ * T_ * SPAD;        // 64 floats

    k_rowlse<<<B_ * T_, 256, 0, stream>>>(predicts, labels, lpext);
    k_ctc  <<<B_,       64,  0, stream>>>(lpext, labels, lens, losses);
    k_mean <<<1,        64,  0, stream>>>(losses, out);
}